// GCNLayer_49246095016424
// MI455X (gfx1250) — compile-verified
//
#include <hip/hip_runtime.h>

typedef __attribute__((ext_vector_type(2))) float v2f;
typedef __attribute__((ext_vector_type(8))) float v8f;

#define N_NODES 50000
#define FDIM    128
#define E_EDGES 1600000

// ---------------------------------------------------------------------------
// Kernel 1: zero the h accumulator in scratch (must re-zero every call).
// ---------------------------------------------------------------------------
__global__ void gcn_zero_kernel(float* __restrict__ h, int n) {
    int i = blockIdx.x * blockDim.x + threadIdx.x;
    if (i < n) h[i] = 0.0f;
}

// ---------------------------------------------------------------------------
// Kernel 2: edge scatter.  One wave32 per edge: lane c reads float4 number c
// of feature[src] (fully coalesced 512B row read, L2-resident) and does 4
// hardware f32 atomics into h[dst].
// ---------------------------------------------------------------------------
__global__ void gcn_scatter_kernel(const float* __restrict__ feature,
                                   const int*   __restrict__ src,
                                   const int*   __restrict__ dst,
                                   float*       __restrict__ h) {
    long long t = (long long)blockIdx.x * blockDim.x + threadIdx.x;
    int e = (int)(t >> 5);
    int c = (int)(t & 31);
    if (e >= E_EDGES) return;

    int s = src[e];
    int d = dst[e];

    const float4* frow = (const float4*)(feature + (size_t)s * FDIM);
    float4 v = frow[c];

    float* hrow = h + (size_t)d * FDIM + (size_t)c * 4;
    unsafeAtomicAdd(hrow + 0, v.x);   // -> global_atomic_add_f32
    unsafeAtomicAdd(hrow + 1, v.y);
    unsafeAtomicAdd(hrow + 2, v.z);
    unsafeAtomicAdd(hrow + 3, v.w);
}

// ---------------------------------------------------------------------------
// Kernel 3: out = h @ W.T + b using V_WMMA_F32_16X16X4_F32 (exact f32 path).
// One wave (32 lanes) computes one 16(M) x 16(N) output tile, K-loop in
// steps of 4 (the f32 WMMA K).
//
// VGPR layouts per CDNA5 ISA 7.12.2 (32-bit operands, wave32):
//   A 16x4 : lane<16 -> {K=k0+0, K=k0+1}; lane>=16 -> {K=k0+2, K=k0+3}; M=lane%16
//   B 4x16 : same K split; N = lane%16;  B[k][n] = W[n][k]
//   C/D    : VGPR v -> M = v + 8*(lane>=16); N = lane%16
// ---------------------------------------------------------------------------
__global__ void gcn_gemm_wmma_kernel(const float* __restrict__ h,
                                     const float* __restrict__ W,
                                     const float* __restrict__ b,
                                     float*       __restrict__ out) {
    const int lane = threadIdx.x;          // blockDim.x == 32
    const int col0 = blockIdx.x * 16;      // 8 tiles across 128 outputs
    const int row0 = blockIdx.y * 16;      // 3125 tiles across 50000 nodes
    const int ln   = lane & 15;
    const int half = lane >> 4;            // 0 or 1
    const int koff = half * 2;

    // Init accumulator with bias: bias depends only on N = ln.
    float bb = b[col0 + ln];
    v8f acc;
#pragma unroll
    for (int v = 0; v < 8; ++v) acc[v] = bb;

    const float* arow = h + (size_t)(row0 + ln) * FDIM;  // A: row M=ln of h-tile
    const float* brow = W + (size_t)(col0 + ln) * FDIM;  // B: B[k][ln] = W[col0+ln][k]

#pragma unroll 4
    for (int k0 = 0; k0 < FDIM; k0 += 4) {
        v2f a, bm;
        a[0]  = arow[k0 + koff];
        a[1]  = arow[k0 + koff + 1];
        bm[0] = brow[k0 + koff];
        bm[1] = brow[k0 + koff + 1];
        // 8 args: (neg_a, A, neg_b, B, c_mod, C, reuse_a, reuse_b)
        acc = __builtin_amdgcn_wmma_f32_16x16x4_f32(false, a, false, bm,
                                                    (short)0, acc, false, false);
    }

#pragma unroll
    for (int v = 0; v < 8; ++v) {
        out[(size_t)(row0 + v + 8 * half) * FDIM + col0 + ln] = acc[v];
    }
}

// ---------------------------------------------------------------------------
// Launcher
// ---------------------------------------------------------------------------
extern "C" void kernel_launch(void* const* d_in, const int* in_sizes, int n_in,
                              void* d_out, int out_size, void* d_ws, size_t ws_size,
                              hipStream_t stream) {
    const float* feature = (const float*)d_in[0];   // [50000, 128] f32
    const int*   src     = (const int*)  d_in[1];   // [1.6M] i32
    const int*   dst     = (const int*)  d_in[2];   // [1.6M] i32
    const float* W       = (const float*)d_in[3];   // [128, 128] f32
    const float* b       = (const float*)d_in[4];   // [128] f32
    float*       out     = (float*)d_out;           // [50000, 128] f32
    float*       h       = (float*)d_ws;            // 25.6 MB accumulator

    // 1) zero h
    int nh = N_NODES * FDIM;
    gcn_zero_kernel<<<(nh + 255) / 256, 256, 0, stream>>>(h, nh);

    // 2) scatter-add: one wave per edge
    long long tot = (long long)E_EDGES * 32;
    unsigned nblk = (unsigned)((tot + 255) / 256);
    gcn_scatter_kernel<<<nblk, 256, 0, stream>>>(feature, src, dst, h);

    // 3) WMMA GEMM: out = h @ W.T + b
    dim3 grid(FDIM / 16, N_NODES / 16);   // (8, 3125)
    gcn_gemm_wmma_kernel<<<grid, 32, 0, stream>>>(h, W, b, out);
}